// SVFIntegration_77781857730827
// MI455X (gfx1250) — compile-verified
//
#include <hip/hip_runtime.h>

// SVF integration by scaling-and-squaring, (1,3,192,192,192) fp32.
// Memory-bound gather kernel: ~1.19 GB total traffic; ping-pong working set
// (2 x 81 MB) fits in the 192 MB L2 -> steps 2..7 mostly L2-resident.
// No matrix structure (data-dependent gather addresses) => no WMMA path;
// optimize data movement: b128 streaming loads/stores, wave32 blocks,
// exec-masked skip of zero-weight trilinear corners, fused 2^-7 scale
// (exact power-of-two => bitwise identical to the reference's scale pass).

#define DD 192
#define HH 192
#define WW 192
constexpr int NVOX = DD * HH * WW;          // 7,077,888 voxels
constexpr int CH   = NVOX;                  // channel stride
constexpr int NQ   = NVOX / 4;              // float4 groups per channel

template <bool SCALED>
__device__ __forceinline__ float ldv(const float* __restrict__ v, int idx) {
    float x = v[idx];
    if (SCALED) x *= (1.0f / 128.0f);       // exact (power of two)
    return x;
}

// One thread handles 4 consecutive x-voxels (same y,z since W%4==0).
// v layout: [c][z][y][x], c-stride = CH.
template <bool SCALED>
__global__ __launch_bounds__(256) void svf_step_kernel(
    const float* __restrict__ v, float* __restrict__ vo)
{
    int t = blockIdx.x * blockDim.x + threadIdx.x;
    if (t >= NQ) return;
    int base = t * 4;
    int x    = base % WW;
    int rem  = base / WW;
    int y    = rem % HH;
    int z    = rem / HH;

    // flow channels: 0 = d(z)-disp, 1 = h(y)-disp, 2 = w(x)-disp
    float4 fd4 = *(const float4*)(v + 0 * CH + base);
    float4 fh4 = *(const float4*)(v + 1 * CH + base);
    float4 fw4 = *(const float4*)(v + 2 * CH + base);
    if (SCALED) {
        const float s = 1.0f / 128.0f;
        fd4.x *= s; fd4.y *= s; fd4.z *= s; fd4.w *= s;
        fh4.x *= s; fh4.y *= s; fh4.z *= s; fh4.w *= s;
        fw4.x *= s; fw4.y *= s; fw4.z *= s; fw4.w *= s;
    }
    float fdv[4] = {fd4.x, fd4.y, fd4.z, fd4.w};
    float fhv[4] = {fh4.x, fh4.y, fh4.z, fh4.w};
    float fwv[4] = {fw4.x, fw4.y, fw4.z, fw4.w};
    float od[4], oh[4], ow[4];

    const float sx = 0.5f * (float)(WW - 1);   // 95.5f
    const float sy = 0.5f * (float)(HH - 1);
    const float sz = 0.5f * (float)(DD - 1);

    #pragma unroll
    for (int j = 0; j < 4; ++j) {
        // Match reference exactly: coords = grid + flow (voxel units), then
        // un-normalized as if in [-1,1]:  i = ((c + 1) * 0.5) * (size-1).
        // (c+1)*0.5 is exact, so folding to (c+1)*95.5 is bitwise identical.
        float ix = ((float)(x + j) + fwv[j] + 1.0f) * sx;
        float iy = ((float)y       + fhv[j] + 1.0f) * sy;
        float iz = ((float)z       + fdv[j] + 1.0f) * sz;

        float fx = floorf(ix), fy = floorf(iy), fz = floorf(iz);
        float tx = ix - fx,   ty = iy - fy,   tz = iz - fz;
        int x0 = (int)fx, y0 = (int)fy, z0 = (int)fz;

        float a0 = 0.0f, a1 = 0.0f, a2 = 0.0f;
        // Reference corner order: x outer, y mid, z inner.
        #pragma unroll
        for (int cx = 0; cx < 2; ++cx) {
            int   xi = x0 + cx;
            float wx = cx ? tx : (1.0f - tx);
            #pragma unroll
            for (int cy = 0; cy < 2; ++cy) {
                int   yi = y0 + cy;
                float wy = cy ? ty : (1.0f - ty);
                #pragma unroll
                for (int cz = 0; cz < 2; ++cz) {
                    int   zi = z0 + cz;
                    float wz = cz ? tz : (1.0f - tz);
                    bool valid = (xi >= 0) & (xi < WW) & (yi >= 0) & (yi < HH)
                               & (zi >= 0) & (zi < DD);
                    float w = valid ? (wx * wy) * wz : 0.0f;
                    // w==0 contributes exactly 0 in the reference (finite
                    // clamped value * 0.0) -> skip gathers under exec mask.
                    if (w != 0.0f) {
                        int idx = (zi * HH + yi) * WW + xi;
                        a0 += ldv<SCALED>(v, idx)          * w;
                        a1 += ldv<SCALED>(v, CH + idx)     * w;
                        a2 += ldv<SCALED>(v, 2 * CH + idx) * w;
                    }
                }
            }
        }
        od[j] = fdv[j] + a0;
        oh[j] = fhv[j] + a1;
        ow[j] = fwv[j] + a2;
    }

    *(float4*)(vo + 0 * CH + base) = make_float4(od[0], od[1], od[2], od[3]);
    *(float4*)(vo + 1 * CH + base) = make_float4(oh[0], oh[1], oh[2], oh[3]);
    *(float4*)(vo + 2 * CH + base) = make_float4(ow[0], ow[1], ow[2], ow[3]);
}

extern "C" void kernel_launch(void* const* d_in, const int* in_sizes, int n_in,
                              void* d_out, int out_size, void* d_ws, size_t ws_size,
                              hipStream_t stream) {
    (void)in_sizes; (void)n_in; (void)out_size; (void)ws_size;
    const float* svf = (const float*)d_in[0];
    float* bufA = (float*)d_ws;    // needs 3*192^3*4 = ~81 MiB of scratch
    float* bufB = (float*)d_out;

    const int blocks = (NQ + 255) / 256;

    // Step 1 fused with the exact 2^-7 scale: reads svf, writes d_out.
    svf_step_kernel<true><<<blocks, 256, 0, stream>>>(svf, bufB);

    // Steps 2..7: strict two-pointer ping-pong B->A->B->A->B->A->B.
    // Even count (6) starting from B lands the final result back in B = d_out.
    const float* cur = bufB;
    float* nxt = bufA;
    for (int s = 0; s < 6; ++s) {
        svf_step_kernel<false><<<blocks, 256, 0, stream>>>(cur, nxt);
        const float* t = cur;
        cur = nxt;
        nxt = (float*)t;
    }
}